// WLConv_11553462026972
// MI455X (gfx1250) — compile-verified
//
#include <hip/hip_runtime.h>

// WL refinement (one round) for MI455X / gfx1250.
// Integer hash + scatter workload: no FP math, so the win comes from the
// memory system (L2-resident atomics, b128 streamed edge reads +
// global_prefetch_b8, LDS mix-LUT) and wave32-sized LDS scans — not from
// WMMA, which cannot express u64 hashing / ordered dedup.

typedef unsigned long long u64;
typedef unsigned int u32;

#define EMPTY_KEY 0xFFFFFFFFFFFFFFFFull
#define LUT_SIZE 64

__device__ __forceinline__ u64 mix64(u64 z) {
    // splitmix64 finalizer (matches the reference's _mix exactly, mod 2^64)
    z += 0x9E3779B97F4A7C15ull;
    z = (z ^ (z >> 30)) * 0xBF58476D1CE4E5B9ull;
    z = (z ^ (z >> 27)) * 0x94D049BB133111EBull;
    return z ^ (z >> 31);
}

// ---- 0) re-initialize all workspace state every call (deterministic) -------
__global__ void k_init(u64* __restrict__ sigbuf, u64* __restrict__ keys,
                       u32* __restrict__ gmin, u32* __restrict__ flags,
                       int N, int M) {
    int i = blockIdx.x * blockDim.x + threadIdx.x;
    if (i < M) { keys[i] = EMPTY_KEY; gmin[i] = 0xFFFFFFFFu; }
    if (i < N) { sigbuf[i] = 0ull;    flags[i] = 0u; }
}

// ---- 1) multiset neighbor hash: neigh[col[e]] += mix(x[row[e]]) ------------
// Vector path: 4 edges/thread via b128 loads (requires E % 4 == 0 so that
// col = ei + E stays 16B-aligned). LDS LUT caches mix64 of small x values.
__global__ void k_edges_vec(const int4* __restrict__ row4, const int4* __restrict__ col4,
                            const int* __restrict__ x, u64* __restrict__ neigh,
                            int nquads) {
    __shared__ u64 lut[LUT_SIZE];
    if (threadIdx.x < LUT_SIZE) lut[threadIdx.x] = mix64((u64)threadIdx.x);
    __syncthreads();

    int q = blockIdx.x * blockDim.x + threadIdx.x;
    if (q >= nquads) return;
#if defined(__gfx1250__)
    // Stream-ahead prefetch of the edge arrays (global_prefetch_b8):
    // keeps HBM reads ahead of the L2 atomic pipeline.
    if (q + 4096 < nquads) {
        __builtin_prefetch(&row4[q + 4096], 0, 1);
        __builtin_prefetch(&col4[q + 4096], 0, 1);
    }
#endif
    int4 r = row4[q];
    int4 c = col4[q];
    u32 x0 = (u32)x[r.x], x1 = (u32)x[r.y], x2 = (u32)x[r.z], x3 = (u32)x[r.w];
    u64 v0 = (x0 < LUT_SIZE) ? lut[x0] : mix64((u64)(long long)(int)x0);
    u64 v1 = (x1 < LUT_SIZE) ? lut[x1] : mix64((u64)(long long)(int)x1);
    u64 v2 = (x2 < LUT_SIZE) ? lut[x2] : mix64((u64)(long long)(int)x2);
    u64 v3 = (x3 < LUT_SIZE) ? lut[x3] : mix64((u64)(long long)(int)x3);
    atomicAdd(&neigh[c.x], v0);   // u64 integer adds: order-independent
    atomicAdd(&neigh[c.y], v1);
    atomicAdd(&neigh[c.z], v2);
    atomicAdd(&neigh[c.w], v3);
}

// Scalar fallback (any E / alignment).
__global__ void k_edges(const int* __restrict__ row, const int* __restrict__ col,
                        const int* __restrict__ x, u64* __restrict__ neigh, int E) {
    int e = blockIdx.x * blockDim.x + threadIdx.x;
    if (e >= E) return;
#if defined(__gfx1250__)
    if (e + 16384 < E) {
        __builtin_prefetch(&row[e + 16384], 0, 1);
        __builtin_prefetch(&col[e + 16384], 0, 1);
    }
#endif
    int r = row[e];
    int c = col[e];
    u64 v = mix64((u64)(long long)x[r]);
    atomicAdd(&neigh[c], v);
}

// ---- 2) per-node signature: sig = mix(own + neigh) (in place over neigh) ---
__global__ void k_sig(const int* __restrict__ x, u64* __restrict__ sigbuf, int N) {
    int i = blockIdx.x * blockDim.x + threadIdx.x;
    if (i >= N) return;
    u64 own = mix64(((u64)(long long)x[i]) ^ 0xABCD1234DEADBEEFull);
    sigbuf[i] = mix64(own + sigbuf[i]);
}

// ---- 3) hash-table dedup: one slot per distinct sig; track min node index --
__global__ void k_insert(const u64* __restrict__ sig, u64* __restrict__ keys,
                         u32* __restrict__ gmin, u32* __restrict__ slot,
                         int N, u32 mask) {
    int i = blockIdx.x * blockDim.x + threadIdx.x;
    if (i >= N) return;
    u64 k = sig[i];                // already well mixed -> direct index
    u32 s = (u32)k & mask;
    for (;;) {
        u64 prev = atomicCAS(&keys[s], EMPTY_KEY, k);
        if (prev == EMPTY_KEY || prev == k) {
            atomicMin(&gmin[s], (u32)i);   // first-occurrence node index
            slot[i] = s;
            return;
        }
        s = (s + 1) & mask;        // linear probe (keys are immutable once set)
    }
}

// ---- 4) mark group leaders: flags[min_index_of_group] = 1 ------------------
__global__ void k_flags(const u64* __restrict__ keys, const u32* __restrict__ gmin,
                        u32* __restrict__ flags, int M) {
    int s = blockIdx.x * blockDim.x + threadIdx.x;
    if (s < M && keys[s] != EMPTY_KEY) flags[gmin[s]] = 1u;
}

// ---- 5a) per-block sums of flags (LDS reduction) ---------------------------
__global__ void k_bsum(const u32* __restrict__ flags, u32* __restrict__ bsum, int N) {
    __shared__ u32 sh[256];
    int i = blockIdx.x * 256 + threadIdx.x;
    sh[threadIdx.x] = (i < N) ? flags[i] : 0u;
    __syncthreads();
    for (int off = 128; off > 0; off >>= 1) {
        if ((int)threadIdx.x < off) sh[threadIdx.x] += sh[threadIdx.x + off];
        __syncthreads();
    }
    if (threadIdx.x == 0) bsum[blockIdx.x] = sh[0];
}

// ---- 5b) exclusive scan of block sums (single 1024-thread block) -----------
__global__ void k_bscan(u32* __restrict__ bsum, int nb) {
    __shared__ u32 sh[1024];
    u32 v = ((int)threadIdx.x < nb) ? bsum[threadIdx.x] : 0u;
    sh[threadIdx.x] = v;
    __syncthreads();
    for (int off = 1; off < 1024; off <<= 1) {
        u32 t = ((int)threadIdx.x >= off) ? sh[threadIdx.x - off] : 0u;
        __syncthreads();
        sh[threadIdx.x] += t;
        __syncthreads();
    }
    if ((int)threadIdx.x < nb) bsum[threadIdx.x] = sh[threadIdx.x] - v; // exclusive
}

// ---- 5c) full exclusive prefix of flags, written in place ------------------
__global__ void k_prefix(const u32* __restrict__ boff, u32* __restrict__ flags, int N) {
    __shared__ u32 sh[256];
    int i = blockIdx.x * 256 + threadIdx.x;
    u32 v = (i < N) ? flags[i] : 0u;
    sh[threadIdx.x] = v;
    __syncthreads();
    for (int off = 1; off < 256; off <<= 1) {
        u32 t = ((int)threadIdx.x >= off) ? sh[threadIdx.x - off] : 0u;
        __syncthreads();
        sh[threadIdx.x] += t;
        __syncthreads();
    }
    if (i < N) flags[i] = sh[threadIdx.x] - v + boff[blockIdx.x];
}

// ---- 6) gather final colors: out[i] = rank(first index of i's group) -------
__global__ void k_out(const u32* __restrict__ slot, const u32* __restrict__ gmin,
                      const u32* __restrict__ prefix, float* __restrict__ out, int N) {
    int i = blockIdx.x * blockDim.x + threadIdx.x;
    if (i >= N) return;
    out[i] = (float)prefix[gmin[slot[i]]];
}

extern "C" void kernel_launch(void* const* d_in, const int* in_sizes, int n_in,
                              void* d_out, int out_size, void* d_ws, size_t ws_size,
                              hipStream_t stream) {
    const int* x  = (const int*)d_in[0];      // (N,)   values 0..49
    const int* ei = (const int*)d_in[1];      // (2,E) flattened row-major
    int N = in_sizes[0];
    int E = in_sizes[1] / 2;
    const int* row = ei;        // senders
    const int* col = ei + E;    // receivers

    // Hash table size: power of two, load factor <= 0.5
    u32 M = 256u;
    while (M < 2u * (u32)N) M <<= 1u;         // N=100000 -> M=262144
    u32 mask = M - 1u;

    // Workspace carve-up (~4.8 MB total)
    char* ws = (char*)d_ws;
    size_t off = 0;
    u64* sigbuf = (u64*)(ws + off); off += ((size_t)N * 8 + 15) & ~(size_t)15;
    u64* keys   = (u64*)(ws + off); off += ((size_t)M * 8 + 15) & ~(size_t)15;
    u32* gmin   = (u32*)(ws + off); off += ((size_t)M * 4 + 15) & ~(size_t)15;
    u32* slot   = (u32*)(ws + off); off += ((size_t)N * 4 + 15) & ~(size_t)15;
    u32* flags  = (u32*)(ws + off); off += ((size_t)N * 4 + 15) & ~(size_t)15;
    u32* bsum   = (u32*)(ws + off); off += 1024 * 4;

    int nbN = (N + 255) / 256;                // 391 blocks (<=1024 for k_bscan)
    int nbM = ((int)M + 255) / 256;
    int nbInit = nbN > nbM ? nbN : nbM;

    k_init<<<nbInit, 256, 0, stream>>>(sigbuf, keys, gmin, flags, N, (int)M);

    if ((E & 3) == 0) {
        // 16B-aligned vector path: 4 edges/thread, b128 loads
        int nquads = E / 4;
        int nbQ = (nquads + 255) / 256;
        k_edges_vec<<<nbQ, 256, 0, stream>>>((const int4*)row, (const int4*)col,
                                             x, sigbuf, nquads);
    } else {
        int nbE = (E + 255) / 256;
        k_edges<<<nbE, 256, 0, stream>>>(row, col, x, sigbuf, E);
    }

    k_sig   <<<nbN, 256, 0, stream>>>(x, sigbuf, N);
    k_insert<<<nbN, 256, 0, stream>>>(sigbuf, keys, gmin, slot, N, mask);
    k_flags <<<nbM, 256, 0, stream>>>(keys, gmin, flags, (int)M);
    k_bsum  <<<nbN, 256, 0, stream>>>(flags, bsum, N);
    k_bscan <<<1,  1024, 0, stream>>>(bsum, nbN);
    k_prefix<<<nbN, 256, 0, stream>>>(bsum, flags, N);
    k_out   <<<nbN, 256, 0, stream>>>(slot, gmin, flags, (float*)d_out, N);
}